// RuzickaKernel_78718160601148
// MI455X (gfx1250) — compile-verified
//
#include <hip/hip_runtime.h>

// Ruzicka similarity: out[i,j] = sum_k min(x[i,k],y[j,k]) / (sum_k max(...) + 1e-8)
// Identity: min+max = a+b  =>  denom = Sx[i] + Sy[j] - num + eps.
// Row sums Sx, Sy computed with V_WMMA_F32_16X16X4_F32 against a ones matrix.
// Main (min,+) loop: 64x64 LDS-tiled, 4x4 register micro-tile, wave32,
// double-buffered async global->LDS staging (gfx1250 ASYNCcnt pipeline).

#define EPS 1e-8f
#define KDIM 256
#define NROWS 1024
#define TILE 64
#define KC 32
#define LSTRIDE 36  // 36 floats/row: 144B (16B aligned) and bank-conflict-free

typedef __attribute__((ext_vector_type(2))) float v2f;
typedef __attribute__((ext_vector_type(8))) float v8f;
typedef __attribute__((ext_vector_type(4))) int v4i;

#define HAS_WMMA_F32X4 __has_builtin(__builtin_amdgcn_wmma_f32_16x16x4_f32)
#define HAS_ASYNC (__has_builtin(__builtin_amdgcn_global_load_async_to_lds_b128) && \
                   __has_builtin(__builtin_amdgcn_s_wait_asynccnt))

// Async builtin takes typed 128-bit vector pointers:
//   param0: v4i addrspace(1)* (global source), param1: v4i addrspace(3)* (LDS dest)
#define AS1V4(p) ((__attribute__((address_space(1))) v4i*)(p))
#define AS3V4(p) ((__attribute__((address_space(3))) v4i*)(p))

// ---------------------------------------------------------------------------
// Row sums via WMMA: one wave per 16 rows. A = 16x4 chunk of the matrix,
// B = 4x16 ones => every column of C accumulates the row sums exactly (f32).
// Blocks 0..63 -> X/Sx, blocks 64..127 -> Y/Sy.
// ---------------------------------------------------------------------------
__global__ __launch_bounds__(32) void rowsum_wmma_kernel(
    const float* __restrict__ X, const float* __restrict__ Y,
    float* __restrict__ Sx, float* __restrict__ Sy) {
  const int b = blockIdx.x;
  const float* __restrict__ M = (b < 64) ? X : Y;
  float* __restrict__ S = (b < 64) ? Sx : Sy;
  const int base = (b & 63) * 16;
#if HAS_WMMA_F32X4
  // 16x4 f32 A layout: lanes 0-15 hold M=lane, VGPR0:{K=0|K=2}, VGPR1:{K=1|K=3}
  const int lane = threadIdx.x & 31;
  const int m = lane & 15;
  const int koff = (lane >> 4) << 1;  // 0 for lanes 0-15, 2 for lanes 16-31
  const float* __restrict__ row = M + (base + m) * KDIM + koff;
  v2f bones = {1.0f, 1.0f};  // ones matrix: layout-independent
  v8f c = {};
#pragma unroll 4
  for (int k = 0; k < KDIM; k += 4) {
    v2f a;
    a.x = row[k];
    a.y = row[k + 1];
    c = __builtin_amdgcn_wmma_f32_16x16x4_f32(false, a, false, bones,
                                              (short)0, c, false, false);
  }
  // C layout: lane L<16 VGPR r = row r col L ; lane L>=16 VGPR r = row r+8.
  if (lane == 0) {
#pragma unroll
    for (int r = 0; r < 8; ++r) S[base + r] = c[r];
  } else if (lane == 16) {
#pragma unroll
    for (int r = 0; r < 8; ++r) S[base + 8 + r] = c[r];
  }
#else
  if (threadIdx.x < 16) {
    float s = 0.0f;
    const float* row = M + (base + threadIdx.x) * KDIM;
    for (int k = 0; k < KDIM; ++k) s += row[k];
    S[base + threadIdx.x] = s;
  }
#endif
}

// ---------------------------------------------------------------------------
// Main (min,+) tile kernel. 256 threads = 8 wave32 per 64x64 output tile.
// Double-buffered: async loads for chunk n+1 are in flight during chunk n's
// VALU work; s_wait_asynccnt(0) + barrier at the stage boundary.
// ---------------------------------------------------------------------------
template <bool USE_SUMS>
__global__ __launch_bounds__(256) void ruzicka_kernel(
    const float* __restrict__ X, const float* __restrict__ Y,
    const float* __restrict__ Sx, const float* __restrict__ Sy,
    float* __restrict__ out) {
  __shared__ float xs[2][TILE * LSTRIDE];
  __shared__ float ys[2][TILE * LSTRIDE];

  const int tid = threadIdx.x;
  const int tx = tid & 15;   // output column group
  const int ty = tid >> 4;   // output row group
  const int tileM = blockIdx.y * TILE;
  const int tileN = blockIdx.x * TILE;

  float accn[4][4];
  float accd[4][4];
#pragma unroll
  for (int i = 0; i < 4; ++i)
#pragma unroll
    for (int j = 0; j < 4; ++j) {
      accn[i][j] = 0.0f;
      accd[i][j] = 0.0f;
    }

  // Loader mapping: 64 rows x 8 float4 columns per matrix; each thread moves
  // two float4 per matrix per K-chunk.
  const int r0 = tid >> 3;         // rows 0..31
  const int r1 = r0 + 32;          // rows 32..63
  const int c4 = (tid & 7) << 2;   // float offset 0,4,...,28 (16B aligned)

  // Stage a K-chunk into LDS buffer `buf` (async on gfx1250).
  auto load_chunk = [&](int buf, int k0) {
#if HAS_ASYNC
    __builtin_amdgcn_global_load_async_to_lds_b128(
        AS1V4(X + (tileM + r0) * KDIM + k0 + c4),
        AS3V4(&xs[buf][r0 * LSTRIDE + c4]), 0, 0);
    __builtin_amdgcn_global_load_async_to_lds_b128(
        AS1V4(X + (tileM + r1) * KDIM + k0 + c4),
        AS3V4(&xs[buf][r1 * LSTRIDE + c4]), 0, 0);
    __builtin_amdgcn_global_load_async_to_lds_b128(
        AS1V4(Y + (tileN + r0) * KDIM + k0 + c4),
        AS3V4(&ys[buf][r0 * LSTRIDE + c4]), 0, 0);
    __builtin_amdgcn_global_load_async_to_lds_b128(
        AS1V4(Y + (tileN + r1) * KDIM + k0 + c4),
        AS3V4(&ys[buf][r1 * LSTRIDE + c4]), 0, 0);
#else
    float4 vx0 = *(const float4*)(X + (tileM + r0) * KDIM + k0 + c4);
    float4 vx1 = *(const float4*)(X + (tileM + r1) * KDIM + k0 + c4);
    float4 vy0 = *(const float4*)(Y + (tileN + r0) * KDIM + k0 + c4);
    float4 vy1 = *(const float4*)(Y + (tileN + r1) * KDIM + k0 + c4);
    *(float4*)(&xs[buf][r0 * LSTRIDE + c4]) = vx0;
    *(float4*)(&xs[buf][r1 * LSTRIDE + c4]) = vx1;
    *(float4*)(&ys[buf][r0 * LSTRIDE + c4]) = vy0;
    *(float4*)(&ys[buf][r1 * LSTRIDE + c4]) = vy1;
#endif
  };

  // Wait for this wave's in-flight async copies, then block-wide barrier.
  auto stage_sync = [&]() {
#if HAS_ASYNC
    __builtin_amdgcn_s_wait_asynccnt(0);
#endif
    __syncthreads();
  };

  // Accumulate one staged K-chunk from LDS buffer `buf`.
  auto compute_chunk = [&](int buf) {
    const float* __restrict__ xsb = &xs[buf][0];
    const float* __restrict__ ysb = &ys[buf][0];
#pragma unroll
    for (int kk = 0; kk < KC; kk += 4) {
      float4 av[4], bv[4];
#pragma unroll
      for (int i = 0; i < 4; ++i)
        av[i] = *(const float4*)(xsb + (ty + 16 * i) * LSTRIDE + kk);
#pragma unroll
      for (int j = 0; j < 4; ++j)
        bv[j] = *(const float4*)(ysb + (tx + 16 * j) * LSTRIDE + kk);
#pragma unroll
      for (int i = 0; i < 4; ++i) {
#pragma unroll
        for (int j = 0; j < 4; ++j) {
          accn[i][j] += fminf(av[i].x, bv[j].x) + fminf(av[i].y, bv[j].y) +
                        fminf(av[i].z, bv[j].z) + fminf(av[i].w, bv[j].w);
          if (!USE_SUMS) {
            accd[i][j] += fmaxf(av[i].x, bv[j].x) + fmaxf(av[i].y, bv[j].y) +
                          fmaxf(av[i].z, bv[j].z) + fmaxf(av[i].w, bv[j].w);
          }
        }
      }
    }
  };

  // Software pipeline: prologue fill, then 2 chunks per iteration so the
  // buffer index is a compile-time constant (no dynamic LDS base math).
  load_chunk(0, 0);
  stage_sync();
  for (int k0 = 0; k0 < KDIM; k0 += 2 * KC) {
    if (k0 + KC < KDIM) load_chunk(1, k0 + KC);
    compute_chunk(0);
    stage_sync();
    if (k0 + KC < KDIM) {
      if (k0 + 2 * KC < KDIM) load_chunk(0, k0 + 2 * KC);
      compute_chunk(1);
      stage_sync();
    }
  }

#pragma unroll
  for (int i = 0; i < 4; ++i) {
    const int row = tileM + ty + 16 * i;
    const float sx = USE_SUMS ? Sx[row] : 0.0f;
#pragma unroll
    for (int j = 0; j < 4; ++j) {
      const int col = tileN + tx + 16 * j;
      const float num = accn[i][j];
      const float den =
          USE_SUMS ? (sx + Sy[col] - num + EPS) : (accd[i][j] + EPS);
      out[row * NROWS + col] = num / den;
    }
  }
}

// ---------------------------------------------------------------------------
extern "C" void kernel_launch(void* const* d_in, const int* in_sizes, int n_in,
                              void* d_out, int out_size, void* d_ws,
                              size_t ws_size, hipStream_t stream) {
  (void)in_sizes;
  (void)n_in;
  (void)out_size;
  const float* X = (const float*)d_in[0];
  const float* Y = (const float*)d_in[1];
  float* out = (float*)d_out;
  dim3 grid(NROWS / TILE, NROWS / TILE);
  if (ws_size >= 2 * NROWS * sizeof(float)) {
    float* Sx = (float*)d_ws;
    float* Sy = Sx + NROWS;
    rowsum_wmma_kernel<<<128, 32, 0, stream>>>(X, Y, Sx, Sy);
    ruzicka_kernel<true><<<grid, 256, 0, stream>>>(X, Y, Sx, Sy, out);
  } else {
    ruzicka_kernel<false><<<grid, 256, 0, stream>>>(X, Y, nullptr, nullptr, out);
  }
}